// LPD_42442866819303
// MI455X (gfx1250) — compile-verified
//
#include <hip/hip_runtime.h>
#include <stdint.h>

#define NPRI   119130
#define NBATCH 16
#define TOPK   2000
#define KEEPK  750
#define NMS_T  0.3f
#define CONF_T 0.3f

typedef unsigned int u32x4 __attribute__((ext_vector_type(4)));
typedef int          i32x8 __attribute__((ext_vector_type(8)));
typedef int          i32x4 __attribute__((ext_vector_type(4)));

// ---------------- Kernel 1: scores = sqrt(sigmoid(c1-c0)*clip(iou,0,1)), thresholded
__global__ void lpd_scores_kernel(const float* __restrict__ conf,
                                  const float* __restrict__ iou,
                                  float* __restrict__ scores, int total) {
  int i = blockIdx.x * blockDim.x + threadIdx.x;
  if (i >= total) return;
  float c0 = conf[2 * i], c1 = conf[2 * i + 1];
  float p1 = 1.0f / (1.0f + expf(c0 - c1));         // softmax(...)[1]
  float u  = fminf(fmaxf(iou[i], 0.0f), 1.0f);
  float s  = sqrtf(p1 * u);
  scores[i] = (s >= CONF_T) ? s : 0.0f;
}

// ---------------- Kernel 2: exact top-2000 per batch (radix-select + bitonic sort)
__device__ __forceinline__ unsigned long long lpd_make_key(float s, int i) {
  // scores >= 0 -> float bits monotone; low 32 bits break ties by smaller index first
  return ((unsigned long long)__float_as_uint(s) << 32) | (unsigned)(NPRI - 1 - i);
}

__global__ __launch_bounds__(1024)
void lpd_select_kernel(const float* __restrict__ scores,
                       int* __restrict__ top_idx, float* __restrict__ top_score) {
  const int b = blockIdx.x, tid = threadIdx.x;
  const float* sc = scores + (size_t)b * NPRI;

  __shared__ unsigned hist[256];
  __shared__ int s_digit, s_remk, s_cnt;
  __shared__ unsigned long long s_keys[2048];

  unsigned long long prefix = 0ull;
  int remk = TOPK;
  for (int pos = 56; pos >= 0; pos -= 8) {
    for (int d = tid; d < 256; d += 1024) hist[d] = 0u;
    __syncthreads();
    for (int i = tid; i < NPRI; i += 1024) {
      unsigned long long key = lpd_make_key(sc[i], i);
      bool cand = (pos == 56) || ((key >> (pos + 8)) == (prefix >> (pos + 8)));
      if (cand) atomicAdd(&hist[(unsigned)(key >> pos) & 255u], 1u);
    }
    __syncthreads();
    if (tid == 0) {
      unsigned cum = 0; int d = 0, rk = remk;
      for (int dd = 255; dd >= 0; --dd) {
        cum += hist[dd];
        if (cum >= (unsigned)rk) { d = dd; rk -= (int)(cum - hist[dd]); break; }
      }
      s_digit = d; s_remk = rk;
    }
    __syncthreads();
    prefix |= ((unsigned long long)(unsigned)s_digit) << pos;
    remk = s_remk;
    __syncthreads();
  }
  // prefix is now exactly the 2000th-largest key; keys are unique -> >= selects 2000
  if (tid == 0) s_cnt = 0;
  for (int i = tid; i < 2048; i += 1024) s_keys[i] = 0ull;
  __syncthreads();
  for (int i = tid; i < NPRI; i += 1024) {
    unsigned long long key = lpd_make_key(sc[i], i);
    if (key >= prefix) {
      int p = atomicAdd(&s_cnt, 1);
      if (p < 2048) s_keys[p] = key;
    }
  }
  __syncthreads();
  // bitonic sort, descending, 2048 elems / 1024 threads
  for (unsigned k = 2; k <= 2048; k <<= 1) {
    for (unsigned j = k >> 1; j > 0; j >>= 1) {
      for (unsigned i = tid; i < 2048; i += 1024) {
        unsigned p = i ^ j;
        if (p > i) {
          unsigned long long a = s_keys[i], c = s_keys[p];
          bool desc = ((i & k) == 0);
          if (desc ? (a < c) : (a > c)) { s_keys[i] = c; s_keys[p] = a; }
        }
      }
      __syncthreads();
    }
  }
  for (int j = tid; j < TOPK; j += 1024) {
    unsigned long long key = s_keys[j];
    top_idx[b * TOPK + j]   = NPRI - 1 - (int)(unsigned)(key & 0xffffffffu);
    top_score[b * TOPK + j] = __uint_as_float((unsigned)(key >> 32));
  }
}

// ---------------- Kernel 3: gather + decode only the 2000 winners (procedural priors)
__global__ void lpd_decode_kernel(const float* __restrict__ loc,
                                  const int* __restrict__ top_idx,
                                  float* __restrict__ out,
                                  float* __restrict__ boxes4) {
  int g = blockIdx.x * blockDim.x + threadIdx.x;
  if (g >= NBATCH * TOPK) return;
  int b = g / TOPK;
  int idx = top_idx[g];

  int base, step, fw, nm; double m0, m1, m2;
  if      (idx <  97200) { base = 0;      step = 8;  fw = 240; nm = 3; m0 = 10;  m1 = 16;  m2 = 24;  }
  else if (idx < 113520) { base = 97200;  step = 16; fw = 120; nm = 2; m0 = 32;  m1 = 48;  m2 = 48;  }
  else if (idx < 117600) { base = 113520; step = 32; fw = 60;  nm = 2; m0 = 64;  m1 = 96;  m2 = 96;  }
  else                   { base = 117600; step = 64; fw = 30;  nm = 3; m0 = 128; m1 = 192; m2 = 256; }
  int r = idx - base;
  int m = r % nm, cell = r / nm;
  int ii = cell / fw, jj = cell - ii * fw;
  double ms = (m == 0) ? m0 : ((m == 1) ? m1 : m2);
  float cx = (float)((jj + 0.5) * step / 1920.0);
  float cy = (float)((ii + 0.5) * step / 1080.0);
  float sw = (float)(ms / 1920.0);
  float sh = (float)(ms / 1080.0);

  const float* L = loc + ((size_t)b * NPRI + idx) * 14;
  float pcx = cx + L[0] * 0.1f * sw;
  float pcy = cy + L[1] * 0.1f * sh;
  float w = sw * expf(L[2] * 0.1f);
  float h = sh * expf(L[3] * 0.2f);
  float x1 = pcx - 0.5f * w, y1 = pcy - 0.5f * h;
  float x2 = x1 + w,         y2 = y1 + h;

  float* O = out + (size_t)g * 15;
  O[0] = x1 * 1920.0f; O[1] = y1 * 1080.0f;
  O[2] = x2 * 1920.0f; O[3] = y2 * 1080.0f;
#pragma unroll
  for (int k = 0; k < 5; ++k) {
    O[4 + 2 * k] = (cx + L[4 + 2 * k] * 0.1f * sw) * 1920.0f;
    O[5 + 2 * k] = (cy + L[5 + 2 * k] * 0.1f * sh) * 1080.0f;
  }
  float* B4 = boxes4 + (size_t)g * 4;
  B4[0] = O[0]; B4[1] = O[1]; B4[2] = O[2]; B4[3] = O[3];
}

// ---------------- Kernel 4: LDS-resident greedy NMS (wave32 chunks), TDM-staged boxes
__device__ __forceinline__ float lpd_iou(float ax1, float ay1, float ax2, float ay2, float aA,
                                         float bx1, float by1, float bx2, float by2, float bA) {
  float iw = fmaxf(fminf(ax2, bx2) - fmaxf(ax1, bx1), 0.0f);
  float ih = fmaxf(fminf(ay2, by2) - fmaxf(ay1, by1), 0.0f);
  float inter = iw * ih;
  return inter / (aA + bA - inter + 1e-9f);
}

__global__ __launch_bounds__(1024)
void lpd_nms_kernel(const float* __restrict__ boxes4,
                    const float* __restrict__ top_score,
                    float* __restrict__ out) {
  const int b = blockIdx.x, tid = threadIdx.x;
  __shared__ __align__(16) float s_box[TOPK * 4];   // 32 KB tile, TDM destination
  __shared__ float s_score[TOPK];
  __shared__ int   s_active[TOPK];
  __shared__ float s_kept[32 * 4];
  __shared__ unsigned s_keptmask;

  // Wave 0 issues a TDM 1D-tile load: 8000 x 4B contiguous -> LDS (ISA ch.8 D#)
  if (tid < 32) {
    unsigned lds_addr = (unsigned)(uintptr_t)(&s_box[0]);   // generic LDS ptr: low 32 bits = LDS offset
    unsigned long long ga = (unsigned long long)(uintptr_t)(boxes4 + (size_t)b * TOPK * 4);
    const unsigned elems = TOPK * 4;                        // 8000 elements of 4 bytes
    u32x4 g0 = {0u, 0u, 0u, 0u};
    g0.x = 1u;                                              // count=1 (valid user descriptor)
    g0.y = lds_addr;                                        // lds_addr [63:32]
    g0.z = (unsigned)(ga & 0xffffffffu);                    // global_addr lo
    g0.w = (unsigned)((ga >> 32) & 0x01ffffffu) | 0x80000000u; // global_addr hi | type=2
    i32x8 g1 = {0, 0, 0, 0, 0, 0, 0, 0};
    g1[0] = (int)(2u << 16);                                // data_size = 4 bytes
    g1[1] = (int)((elems & 0xffffu) << 16);                 // tensor_dim0 lo16 (bits 63:48)
    g1[2] = (int)((elems >> 16) | (1u << 16));              // tensor_dim0 hi16 | tensor_dim1=1
    g1[3] = (int)((elems & 0xffffu) << 16);                 // tile_dim0 = 8000 (bits 127:112)
    g1[4] = 1;                                              // tile_dim1 = 1, tile_dim2 = 0
    g1[5] = (int)elems;                                     // tensor_dim0_stride lo32
    i32x4 g2 = {1, 1, 0, 0};                                // tensor_dim2=1, tensor_dim3=1
    i32x4 g3 = {0, 0, 0, 0};
    i32x8 g4 = {0, 0, 0, 0, 0, 0, 0, 0};                    // extra group (clang-23 6-arg form)
    __builtin_amdgcn_tensor_load_to_lds(g0, g1, g2, g3, g4, 0);
    __builtin_amdgcn_s_wait_tensorcnt(0);
  }
  for (int j = tid; j < TOPK; j += 1024) {
    float s = top_score[b * TOPK + j];
    s_score[j]  = s;
    s_active[j] = (s > 0.0f) ? 1 : 0;                       // reference: active = (s > 0)
  }
  __syncthreads();

  for (int c0 = 0; c0 < TOPK; c0 += 32) {
    if (tid < 32) {                                         // intra-chunk greedy in one wave32
      int i = c0 + (int)tid;
      float x1 = 0.f, y1 = 0.f, x2 = 0.f, y2 = 0.f; int act = 0;
      if (i < TOPK) {
        x1 = s_box[i * 4 + 0]; y1 = s_box[i * 4 + 1];
        x2 = s_box[i * 4 + 2]; y2 = s_box[i * 4 + 3];
        act = s_active[i];
      }
      float A = fmaxf(x2 - x1, 0.f) * fmaxf(y2 - y1, 0.f);
      for (int t = 0; t < 32; ++t) {
        int ta = __shfl(act, t, 32);                        // uniform broadcast
        if (ta) {
          float tx1 = __shfl(x1, t, 32), ty1 = __shfl(y1, t, 32);
          float tx2 = __shfl(x2, t, 32), ty2 = __shfl(y2, t, 32);
          float tA  = __shfl(A,  t, 32);
          if (((int)tid > t) && act) {
            if (lpd_iou(tx1, ty1, tx2, ty2, tA, x1, y1, x2, y2, A) > NMS_T) act = 0;
          }
        }
      }
      if (i < TOPK) s_active[i] = act;
      s_kept[tid * 4 + 0] = x1; s_kept[tid * 4 + 1] = y1;
      s_kept[tid * 4 + 2] = x2; s_kept[tid * 4 + 3] = y2;
      unsigned long long bal = __ballot(act != 0);
      if (tid == 0) s_keptmask = (unsigned)bal;
    }
    __syncthreads();
    unsigned km = s_keptmask;
    if (km) {                                               // suppress the tail in parallel
      for (int j = c0 + 32 + (int)tid; j < TOPK; j += 1024) {
        if (s_active[j]) {
          float jx1 = s_box[j * 4 + 0], jy1 = s_box[j * 4 + 1];
          float jx2 = s_box[j * 4 + 2], jy2 = s_box[j * 4 + 3];
          float jA = fmaxf(jx2 - jx1, 0.f) * fmaxf(jy2 - jy1, 0.f);
          unsigned mm = km; bool dead = false;
          while (mm) {
            int t = __builtin_ctz(mm); mm &= mm - 1;
            float kx1 = s_kept[t * 4 + 0], ky1 = s_kept[t * 4 + 1];
            float kx2 = s_kept[t * 4 + 2], ky2 = s_kept[t * 4 + 3];
            float kA = fmaxf(kx2 - kx1, 0.f) * fmaxf(ky2 - ky1, 0.f);
            if (lpd_iou(kx1, ky1, kx2, ky2, kA, jx1, jy1, jx2, jy2, jA) > NMS_T) { dead = true; break; }
          }
          if (dead) s_active[j] = 0;
        }
      }
    }
    __syncthreads();
  }

  if (tid == 0) {                                           // cumsum(keep) <= 750 cap
    int c = 0;
    for (int j = 0; j < TOPK; ++j)
      if (s_active[j]) { ++c; if (c > KEEPK) s_active[j] = 0; }
  }
  __syncthreads();
  for (int j = tid; j < TOPK; j += 1024)
    out[((size_t)b * TOPK + j) * 15 + 14] = s_active[j] ? s_score[j] : 0.0f;
}

// ---------------- host entry
extern "C" void kernel_launch(void* const* d_in, const int* in_sizes, int n_in,
                              void* d_out, int out_size, void* d_ws, size_t ws_size,
                              hipStream_t stream) {
  const float* loc  = (const float*)d_in[0];   // [16, 119130, 14]
  const float* conf = (const float*)d_in[1];   // [16, 119130, 2]
  const float* iou  = (const float*)d_in[2];   // [16, 119130, 1]
  float* out = (float*)d_out;                  // [16, 2000, 15]

  char* ws = (char*)d_ws;                      // ~8.4 MB total
  float* scores    = (float*)ws;
  size_t off       = (size_t)NBATCH * NPRI * sizeof(float);
  int*   top_idx   = (int*)(ws + off);          off += (size_t)NBATCH * TOPK * sizeof(int);
  float* top_score = (float*)(ws + off);        off += (size_t)NBATCH * TOPK * sizeof(float);
  float* boxes4    = (float*)(ws + off);

  int total = NBATCH * NPRI;
  lpd_scores_kernel<<<(total + 255) / 256, 256, 0, stream>>>(conf, iou, scores, total);
  lpd_select_kernel<<<NBATCH, 1024, 0, stream>>>(scores, top_idx, top_score);
  lpd_decode_kernel<<<(NBATCH * TOPK + 127) / 128, 128, 0, stream>>>(loc, top_idx, out, boxes4);
  lpd_nms_kernel<<<NBATCH, 1024, 0, stream>>>(boxes4, top_score, out);
}